// GNNModel_31576599560907
// MI455X (gfx1250) — compile-verified
//
#include <hip/hip_runtime.h>
#include <stdint.h>

typedef __attribute__((ext_vector_type(2))) float v2f;
typedef __attribute__((ext_vector_type(8))) float v8f;

#define N_NODES 100000
#define NTILES  (N_NODES / 16)   // 6250, N divides 16 exactly

// ---------------------------------------------------------------- degree ----
__global__ void k_init_deg(float* __restrict__ deg, int n) {
  int i = blockIdx.x * blockDim.x + threadIdx.x;
  if (i < n) deg[i] = 1.0f;  // self-loop contributes 1 to every in-degree
}

__global__ void k_deg_scatter(const int* __restrict__ ei, int n_edges,
                              float* __restrict__ deg) {
  int e = blockIdx.x * blockDim.x + threadIdx.x;
  if (e < n_edges) atomicAdd(&deg[ei[n_edges + e]], 1.0f);  // dst row
}

__global__ void k_rsqrt(float* __restrict__ deg_dis, int n) {
  int i = blockIdx.x * blockDim.x + threadIdx.x;
  if (i < n) deg_dis[i] = rsqrtf(deg_dis[i]);  // deg >= 1 always (self-loop)
}

// ---------------------------------------------- layer 1: hs1 = dis*(x@W1) ---
// One wave32 computes a 16x16 output tile via two V_WMMA_F32_16X16X4_F32
// (K = 5 padded to 8). A-layout (16x4 f32): lane L holds row M=L%16;
// VGPR0 = K=(L<16?0:2), VGPR1 = K=(L<16?1:3). C/D: VGPR r -> row r (+8 for
// lanes>=16), col = L%16. Stores both hs1 and the accumulator seed (self-loop).
__global__ void k_layer1_wmma(const float* __restrict__ x,     // [N,5]
                              const float* __restrict__ W1,    // [5,16]
                              const float* __restrict__ dis,   // [N]
                              float* __restrict__ hs1,         // [N,16]
                              float* __restrict__ acc1,        // [N,16]
                              int n_tiles) {
  const int lane = threadIdx.x & 31;
  const int wave = threadIdx.x >> 5;
  const int tile = blockIdx.x * (blockDim.x >> 5) + wave;
  if (tile >= n_tiles) return;

  const int  col = lane & 15;
  const bool hi  = lane >= 16;
  const int  node = tile * 16 + col;      // this lane's A row
  const int  k0   = hi ? 2 : 0;

  v2f a0, a1, b0, b1;
  // K-step 0: k = {0,1} (low half) / {2,3} (high half)
  a0.x = x[node * 5 + k0];
  a0.y = x[node * 5 + k0 + 1];
  b0.x = W1[k0 * 16 + col];
  b0.y = W1[(k0 + 1) * 16 + col];
  // K-step 1: k = {4, pad} (low half) / {pad, pad} (high half)
  a1.x = hi ? 0.0f : x[node * 5 + 4];
  a1.y = 0.0f;
  b1.x = hi ? 0.0f : W1[4 * 16 + col];
  b1.y = 0.0f;

  v8f c = {};
  c = __builtin_amdgcn_wmma_f32_16x16x4_f32(false, a0, false, b0,
                                            (short)0, c, false, false);
  c = __builtin_amdgcn_wmma_f32_16x16x4_f32(false, a1, false, b1,
                                            (short)0, c, false, false);

#pragma unroll
  for (int r = 0; r < 8; ++r) {
    const int m    = r + (hi ? 8 : 0);
    const int nrow = tile * 16 + m;
    const float v  = c[r] * dis[nrow];   // pre-scale by dis[src]
    hs1[nrow * 16 + col]  = v;           // read-only gather source
    acc1[nrow * 16 + col] = v;           // seed = self-loop contribution
  }
}

// --------------------------------------- layer 1 edge scatter (16 ch/edge) --
__global__ void k_scatter1(const int* __restrict__ ei, int n_edges,
                           const float* __restrict__ hs1,
                           float* __restrict__ acc1) {
  const long long t = (long long)blockIdx.x * blockDim.x + threadIdx.x;
  if (t >= (long long)n_edges * 16) return;
  const int e = (int)(t >> 4);
  const int c = (int)(t & 15);
  const int s = ei[e];
  const int d = ei[n_edges + e];
  atomicAdd(&acc1[d * 16 + c], hs1[s * 16 + c]);  // coalesced 64B per half-wave
}

// -------- layer 2 node pass: relu(dis*acc1+b1) @ W2, pre-scaled, seed out ---
__global__ void k_layer2_node(const float* __restrict__ acc1,  // [N,16]
                              const float* __restrict__ dis,   // [N]
                              const float* __restrict__ b1,    // [16]
                              const float* __restrict__ W2,    // [16]
                              float* __restrict__ hs2,         // [N]
                              float* __restrict__ acc2,        // [N] (= d_out)
                              int n) {
  const int i = blockIdx.x * blockDim.x + threadIdx.x;
  if (i >= n) return;
  const float d = dis[i];
  float g = 0.0f;
#pragma unroll
  for (int c = 0; c < 16; ++c) {
    const float hrelu = fmaxf(fmaf(d, acc1[i * 16 + c], b1[c]), 0.0f);
    g = fmaf(hrelu, W2[c], g);
  }
  const float s = d * g;   // pre-scale by dis[src]
  hs2[i]  = s;
  acc2[i] = s;             // seed = self-loop contribution
}

// ------------------------------------------------- layer 2 edge scatter -----
__global__ void k_scatter2(const int* __restrict__ ei, int n_edges,
                           const float* __restrict__ hs2,
                           float* __restrict__ acc2) {
  const int e = blockIdx.x * blockDim.x + threadIdx.x;
  if (e >= n_edges) return;
  atomicAdd(&acc2[ei[n_edges + e]], hs2[ei[e]]);
}

// -------------------------------------------- final: out = dis*acc2 + b2 ----
__global__ void k_final(float* __restrict__ out,               // in-place acc2
                        const float* __restrict__ dis,
                        const float* __restrict__ b2, int n) {
  const int i = blockIdx.x * blockDim.x + threadIdx.x;
  if (i < n) out[i] = fmaf(dis[i], out[i], b2[0]);
}

// ---------------------------------------------------------------------------
extern "C" void kernel_launch(void* const* d_in, const int* in_sizes, int n_in,
                              void* d_out, int out_size, void* d_ws, size_t ws_size,
                              hipStream_t stream) {
  const float* x  = (const float*)d_in[0];   // [N,5]
  const int*   ei = (const int*)  d_in[1];   // [2,E] (harness delivers int32)
  const float* W1 = (const float*)d_in[2];   // [5,16]
  const float* b1 = (const float*)d_in[3];   // [16]
  const float* W2 = (const float*)d_in[4];   // [16,1]
  const float* b2 = (const float*)d_in[5];   // [1]
  float* out = (float*)d_out;                // [N,1], doubles as acc2

  const int n_edges = in_sizes[1] / 2;

  // workspace layout (floats): dis[N] | hs1[16N] | acc1[16N] | hs2[N]  = 34N
  float* ws   = (float*)d_ws;
  float* dis  = ws;
  float* hs1  = ws + (size_t)N_NODES;
  float* acc1 = ws + (size_t)17 * N_NODES;
  float* hs2  = ws + (size_t)33 * N_NODES;

  const int B = 256;  // 8 waves per block on wave32
  const int gN = (N_NODES + B - 1) / B;
  const int gE = (n_edges + B - 1) / B;
  const long long sc1 = (long long)n_edges * 16;
  const int gS1 = (int)((sc1 + B - 1) / B);
  const int gT = (NTILES + 7) / 8;    // 8 waves (tiles) per block

  k_init_deg   <<<gN,  B, 0, stream>>>(dis, N_NODES);
  k_deg_scatter<<<gE,  B, 0, stream>>>(ei, n_edges, dis);
  k_rsqrt      <<<gN,  B, 0, stream>>>(dis, N_NODES);
  k_layer1_wmma<<<gT,  B, 0, stream>>>(x, W1, dis, hs1, acc1, NTILES);
  k_scatter1   <<<gS1, B, 0, stream>>>(ei, n_edges, hs1, acc1);
  k_layer2_node<<<gN,  B, 0, stream>>>(acc1, dis, b1, W2, hs2, out, N_NODES);
  k_scatter2   <<<gE,  B, 0, stream>>>(ei, n_edges, hs2, out);
  k_final      <<<gN,  B, 0, stream>>>(out, dis, b2, N_NODES);
}